// AttentionCl_3590592659726
// MI455X (gfx1250) — compile-verified
//
#include <hip/hip_runtime.h>
#include <hip/hip_bf16.h>

// fp32 WMMA path: V_WMMA_F32_16X16X4_F32  (A=16x4 f32 -> v2f, B=4x16 f32 -> v2f, C/D=16x16 f32 -> v8f)
typedef float v2f __attribute__((ext_vector_type(2)));
typedef float v8f __attribute__((ext_vector_type(8)));

#define WMMA_F32(a, b, c) \
  __builtin_amdgcn_wmma_f32_16x16x4_f32(false, (a), false, (b), (short)0, (c), false, false)

// Problem constants
#define NB      8
#define NSEQ    1024      // H*W
#define NC      512       // channels
#define NHEADS  16
#define DH      32
#define QKVN    1536      // 3*NHEADS*DH

// ---------------------------------------------------------------------------
// Kernel 1: qkv = x @ W_qkv + b_qkv, scattered into head-major Q, K^T, V.
//   Q  : [128][1024][32]   (bh, n, d)
//   Kt : [128][32][1024]   (bh, d, m)   <- transposed so QK^T B-tiles load naturally
//   V  : [128][1024][32]   (bh, m, d)
// Register-blocked: one wave computes a 64x16 output strip (4 M-tiles sharing
// each B fragment), so each v_wmma_f32_16x16x4 costs 1.5 VMEM instrs instead of 3.
// ---------------------------------------------------------------------------
__global__ void qkv_kernel(const float* __restrict__ x,
                           const float* __restrict__ Wqkv,
                           const float* __restrict__ bqkv,
                           float* __restrict__ Qs,
                           float* __restrict__ Kt,
                           float* __restrict__ Vs) {
  const int wave = blockIdx.x * (blockDim.x >> 5) + (threadIdx.x >> 5);
  const int lane = threadIdx.x & 31;
  const int lm = lane & 15;
  const int hi = lane >> 4;
  const int NT = QKVN / 16;                 // 96 tiles along output cols
  const int mt4 = wave / NT;                // 128 strips of 64 rows
  const int nt  = wave % NT;
  if (mt4 >= (NB * NSEQ) / 64) return;      // wave-uniform guard

  const float* Arow[4];
#pragma unroll
  for (int i = 0; i < 4; ++i)
    Arow[i] = x + (size_t)(mt4 * 64 + i * 16 + lm) * NC;   // A row = lane%16
  const int col0 = nt * 16;

  v8f c[4];
#pragma unroll
  for (int i = 0; i < 4; ++i) c[i] = (v8f){0.f, 0.f, 0.f, 0.f, 0.f, 0.f, 0.f, 0.f};

#pragma unroll 2
  for (int k0 = 0; k0 < NC; k0 += 4) {
    v2f b;                                                  // shared B fragment
    b.x = Wqkv[(size_t)(k0 + 2 * hi) * QKVN + col0 + lm];
    b.y = Wqkv[(size_t)(k0 + 2 * hi + 1) * QKVN + col0 + lm];
#pragma unroll
    for (int i = 0; i < 4; ++i) {
      v2f a = *(const v2f*)(Arow[i] + k0 + 2 * hi);         // A[m, k0+2hi .. +1]
      c[i] = WMMA_F32(a, b, c[i]);
    }
  }

#pragma unroll
  for (int i = 0; i < 4; ++i) {
#pragma unroll
    for (int r = 0; r < 8; ++r) {
      const int row = r + hi * 8;            // C/D layout: vgpr r -> rows r / r+8
      const int gm  = mt4 * 64 + i * 16 + row;
      const int col = col0 + lm;
      const float v = c[i][r] + bqkv[col];
      const int head = col / 96;
      const int rr   = col % 96;             // [q(32) | k(32) | v(32)] per head
      const int bb   = gm >> 10;
      const int n    = gm & 1023;
      const int bh   = bb * NHEADS + head;
      if (rr < 32)
        Qs[(size_t)bh * (NSEQ * DH) + n * DH + rr] = v;
      else if (rr < 64)
        Kt[(size_t)bh * (NSEQ * DH) + (rr - 32) * NSEQ + n] = v;
      else
        Vs[(size_t)bh * (NSEQ * DH) + n * DH + (rr - 64)] = v;
    }
  }
}

// ---------------------------------------------------------------------------
// Kernel 2: attention with online (flash) softmax, fp32 WMMA for QK^T and PV.
// One wave owns 16 query rows of one (b,h). 64 key tiles streamed; S tile is
// bounced through LDS to transpose C-layout -> A-layout and compute row stats.
// ---------------------------------------------------------------------------
__global__ void attn_kernel(const float* __restrict__ Qs,
                            const float* __restrict__ Kt,
                            const float* __restrict__ Vs,
                            const float* __restrict__ rel,
                            float* __restrict__ ctx) {
  __shared__ __align__(16) float Pl[8][16 * 16];   // per-wave S/P tile
  __shared__ float Sm[8][16];                      // running row max
  __shared__ float Sl[8][16];                      // running row sum
  __shared__ float Sc[8][16];                      // rescale factor this tile

  const int w    = threadIdx.x >> 5;
  const int lane = threadIdx.x & 31;
  const int lm   = lane & 15;
  const int hi   = lane >> 4;
  const int wave = blockIdx.x * (blockDim.x >> 5) + w;
  const int bh   = wave >> 6;                      // 64 query tiles per (b,h)
  const int nt   = wave & 63;
  if (bh >= NB * NHEADS) return;                   // wave-uniform
  const int head = bh & (NHEADS - 1);
  const int bb   = bh >> 4;
  const int n0   = nt * 16;
  const float scale = 0.17677669529663687f;        // dh^-0.5

  const float* Qb = Qs + (size_t)bh * (NSEQ * DH);
  const float* Kb = Kt + (size_t)bh * (NSEQ * DH);
  const float* Vb = Vs + (size_t)bh * (NSEQ * DH);
  const float* Rb = rel + (size_t)head * NSEQ * NSEQ;

  // Q tile in A-layout registers: 8 chunks of 16x4
  v2f qa[8];
#pragma unroll
  for (int dc = 0; dc < 8; ++dc)
    qa[dc] = *(const v2f*)(Qb + (size_t)(n0 + lm) * DH + dc * 4 + 2 * hi);

  v8f c0 = {0.f, 0.f, 0.f, 0.f, 0.f, 0.f, 0.f, 0.f};  // out cols 0..15
  v8f c1 = {0.f, 0.f, 0.f, 0.f, 0.f, 0.f, 0.f, 0.f};  // out cols 16..31
  if (lane < 16) { Sm[w][lane] = -3.0e38f; Sl[w][lane] = 0.f; }
  float* P = Pl[w];

  for (int mt = 0; mt < 64; ++mt) {
    const int m0 = mt * 16;
    // warm next bias tile in L2 (global_prefetch_b8)
    if (mt + 1 < 64)
      __builtin_prefetch(Rb + (size_t)(n0 + lm) * NSEQ + m0 + 16, 0, 1);

    // ---- S = (Q K^T) tile : 8 x wmma over d ----
    v8f s = {0.f, 0.f, 0.f, 0.f, 0.f, 0.f, 0.f, 0.f};
#pragma unroll
    for (int dc = 0; dc < 8; ++dc) {
      const float* kp = Kb + (size_t)(dc * 4 + 2 * hi) * NSEQ + m0 + lm;
      v2f kb;
      kb.x = kp[0];
      kb.y = kp[NSEQ];
      s = WMMA_F32(qa[dc], kb, s);
    }

    // ---- scale + rel-pos bias, spill tile to LDS ----
#pragma unroll
    for (int r = 0; r < 8; ++r) {
      const int row = r + hi * 8;
      const float sv = s[r] * scale + Rb[(size_t)(n0 + row) * NSEQ + m0 + lm];
      P[row * 16 + lm] = sv;
    }
    asm volatile("s_wait_dscnt 0x0" ::: "memory");

    // ---- per-row online softmax stats (lanes 0..15, row = lane) ----
    if (lane < 16) {
      const int row = lm;
      float tmax = -3.0e38f;
#pragma unroll
      for (int j = 0; j < 16; ++j) tmax = fmaxf(tmax, P[row * 16 + j]);
      const float om = Sm[w][row];
      const float ol = Sl[w][row];
      const float nm = fmaxf(om, tmax);
      const float corr = __expf(om - nm);
      float ps = 0.f;
#pragma unroll
      for (int j = 0; j < 16; ++j) {
        const float e = __expf(P[row * 16 + j] - nm);
        ps += e;
        P[row * 16 + j] = e;                 // P tile now holds exp()
      }
      Sm[w][row] = nm;
      Sl[w][row] = ol * corr + ps;
      Sc[w][row] = corr;
    }
    asm volatile("s_wait_dscnt 0x0" ::: "memory");

    // ---- rescale accumulators by exp(old_max - new_max) ----
#pragma unroll
    for (int r = 0; r < 8; ++r) {
      const float corr = Sc[w][r + hi * 8];
      c0[r] *= corr;
      c1[r] *= corr;
    }

    // ---- O += P V : 4 m-chunks x 2 d-halves of wmma ----
#pragma unroll
    for (int mc = 0; mc < 4; ++mc) {
      v2f pa = *(const v2f*)(P + lm * 16 + mc * 4 + 2 * hi);   // A = P[n, m-chunk]
      const float* vp = Vb + (size_t)(m0 + mc * 4 + 2 * hi) * DH;
      v2f b0, b1;
      b0.x = vp[lm];            b0.y = vp[DH + lm];            // d cols 0..15
      b1.x = vp[16 + lm];       b1.y = vp[DH + 16 + lm];       // d cols 16..31
      c0 = WMMA_F32(pa, b0, c0);
      c1 = WMMA_F32(pa, b1, c1);
    }
  }

  // ---- normalize and write ctx[b, n, head*32 + d] ----
  asm volatile("s_wait_dscnt 0x0" ::: "memory");
#pragma unroll
  for (int r = 0; r < 8; ++r) {
    const int row = r + hi * 8;
    const float inv = 1.0f / Sl[w][row];
    const size_t base = (size_t)(bb * NSEQ + n0 + row) * NC + head * DH;
    ctx[base + lm]      = c0[r] * inv;
    ctx[base + 16 + lm] = c1[r] * inv;
  }
}

// ---------------------------------------------------------------------------
// Kernel 3: out = ctx @ W_proj + b_proj   ([8192x512] x [512x512])
// Same 4x1 register blocking as kernel 1.
// ---------------------------------------------------------------------------
__global__ void proj_kernel(const float* __restrict__ ctx,
                            const float* __restrict__ Wp,
                            const float* __restrict__ bp,
                            float* __restrict__ out) {
  const int wave = blockIdx.x * (blockDim.x >> 5) + (threadIdx.x >> 5);
  const int lane = threadIdx.x & 31;
  const int lm = lane & 15;
  const int hi = lane >> 4;
  const int NT = NC / 16;                    // 32
  const int mt4 = wave / NT;                 // 128 strips of 64 rows
  const int nt  = wave % NT;
  if (mt4 >= (NB * NSEQ) / 64) return;

  const float* Arow[4];
#pragma unroll
  for (int i = 0; i < 4; ++i)
    Arow[i] = ctx + (size_t)(mt4 * 64 + i * 16 + lm) * NC;
  const int col0 = nt * 16;

  v8f c[4];
#pragma unroll
  for (int i = 0; i < 4; ++i) c[i] = (v8f){0.f, 0.f, 0.f, 0.f, 0.f, 0.f, 0.f, 0.f};

#pragma unroll 2
  for (int k0 = 0; k0 < NC; k0 += 4) {
    v2f b;
    b.x = Wp[(size_t)(k0 + 2 * hi) * NC + col0 + lm];
    b.y = Wp[(size_t)(k0 + 2 * hi + 1) * NC + col0 + lm];
#pragma unroll
    for (int i = 0; i < 4; ++i) {
      v2f a = *(const v2f*)(Arow[i] + k0 + 2 * hi);
      c[i] = WMMA_F32(a, b, c[i]);
    }
  }

#pragma unroll
  for (int i = 0; i < 4; ++i) {
#pragma unroll
    for (int r = 0; r < 8; ++r) {
      const int row = r + hi * 8;
      const int gm  = mt4 * 64 + i * 16 + row;
      const int col = col0 + lm;
      out[(size_t)gm * NC + col] = c[i][r] + bp[col];
    }
  }
}

// ---------------------------------------------------------------------------
extern "C" void kernel_launch(void* const* d_in, const int* in_sizes, int n_in,
                              void* d_out, int out_size, void* d_ws, size_t ws_size,
                              hipStream_t stream) {
  const float* x     = (const float*)d_in[0];   // [8,32,32,512]
  const float* rel   = (const float*)d_in[1];   // [16,1024,1024]
  const float* Wqkv  = (const float*)d_in[2];   // [512,1536]
  const float* bqkv  = (const float*)d_in[3];   // [1536]
  const float* Wproj = (const float*)d_in[4];   // [512,512]
  const float* bproj = (const float*)d_in[5];   // [512]
  float* out = (float*)d_out;

  // workspace: Q | Kt | V | ctx, 16 MB each (64 MB total)
  float* ws  = (float*)d_ws;
  float* Qs  = ws;
  float* Kt  = ws + (size_t)4194304;
  float* Vs  = ws + (size_t)8388608;
  float* ctx = ws + (size_t)12582912;

  // 128 strips x 96 col-tiles = 12288 waves / 8 per 256-thread block
  qkv_kernel<<<1536, 256, 0, stream>>>(x, Wqkv, bqkv, Qs, Kt, Vs);
  // 128 (b,h) x 64 query tiles = 8192 waves
  attn_kernel<<<1024, 256, 0, stream>>>(Qs, Kt, Vs, rel, ctx);
  // 128 strips x 32 col-tiles = 4096 waves
  proj_kernel<<<512, 256, 0, stream>>>(ctx, Wproj, bproj, out);
}